// DagnabbitAutoEncoder_6425271074832
// MI455X (gfx1250) — compile-verified
//
#include <hip/hip_runtime.h>

typedef __attribute__((ext_vector_type(2))) float v2f;
typedef __attribute__((ext_vector_type(8))) float v8f;

#define N_TRUNK  65536
#define DIM      128
#define TWOD     256
#define NROOT    64
#define NWG      128
#define NTHREADS 256

// workspace layout (all re-initialized every launch):
//   [0, 256KB)            : done flags, one u32 per trunk node
//   [256KB, 256KB+1MB)    : packed W1  (4 types x 16 tiles x 64 ksteps x 32 lanes x 2)
//   [+1MB, +1.5MB)        : packed W2  (4 types x  8 tiles x 64 ksteps x 32 lanes x 2)
#define FLAGS_BYTES (N_TRUNK * 4)
#define PK1_ELEMS   (4 * 16 * 64 * 32 * 2)   // 262144 floats
#define PK2_ELEMS   (4 * 8  * 64 * 32 * 2)   // 131072 floats

// Branch-free exact-GELU: 0.5*x*(1+erf(x/sqrt(2))) with Abramowitz-Stegun
// 7.1.26 erf (|abs err| <= 1.5e-7). Uses v_rcp_f32 + v_exp_f32 directly so no
// EXEC-mask branching appears on the node critical path.
__device__ __forceinline__ float gelu_exact(float x) {
    const float z = __builtin_fabsf(x) * 0.70710678118654752f;  // |x|/sqrt(2)
    const float t = __builtin_amdgcn_rcpf(__builtin_fmaf(0.3275911f, z, 1.0f));
    const float poly = t * __builtin_fmaf(t,
                           __builtin_fmaf(t,
                           __builtin_fmaf(t,
                           __builtin_fmaf(t, 1.061405429f, -1.453152027f),
                                             1.421413741f),
                                            -0.284496736f),
                                             0.254829592f);
    // exp(-z^2) = exp2(-z^2 * log2(e)); raw v_exp_f32 underflows to 0 for
    // large z, where erf correctly saturates to 1.
    const float e  = __builtin_amdgcn_exp2f(-z * z * 1.4426950408889634f);
    const float erf_abs = __builtin_fmaf(-poly, e, 1.0f);
    const float erf_s   = __builtin_copysignf(erf_abs, x);
    return 0.5f * x * (1.0f + erf_s);
}

// Per-launch init: zero flags, copy roots into out, and pack W1/W2 into the
// exact WMMA B-operand register layout so the hot loop does one coalesced
// global_load_b64 per tile per K-step.
//   B layout (f32 16x16x4): lane L, dword j -> K = 4*kstep + 2*(L>>4) + j,
//                           N = 16*ntile + (L&15)
__global__ void dag_init(const float* __restrict__ roots,
                         const float* __restrict__ W1,
                         const float* __restrict__ W2,
                         float* __restrict__ out,
                         unsigned int* __restrict__ flags,
                         float* __restrict__ pk1,
                         float* __restrict__ pk2) {
    int tid = blockIdx.x * blockDim.x + threadIdx.x;
    if (tid < N_TRUNK) flags[tid] = 0u;
    if (tid < NROOT * DIM) out[tid] = roots[tid];
    if (tid < PK1_ELEMS) {
        int e = tid;
        int j     = e & 1;   e >>= 1;
        int lane  = e & 31;  e >>= 5;
        int kstep = e & 63;  e >>= 6;
        int ntile = e & 15;  e >>= 4;
        int t     = e;       // 0..3
        int k   = 4 * kstep + 2 * (lane >> 4) + j;
        int col = 16 * ntile + (lane & 15);
        pk1[tid] = W1[((size_t)t * TWOD + k) * TWOD + col];
    }
    if (tid < PK2_ELEMS) {
        int e = tid;
        int j     = e & 1;   e >>= 1;
        int lane  = e & 31;  e >>= 5;
        int kstep = e & 63;  e >>= 6;
        int ntile = e & 7;   e >>= 3;
        int t     = e;       // 0..3
        int k   = 4 * kstep + 2 * (lane >> 4) + j;
        int col = 16 * ntile + (lane & 15);
        pk2[tid] = W2[((size_t)t * TWOD + k) * DIM + col];
    }
}

__global__ __launch_bounds__(NTHREADS) void dag_mlp(
    const int* __restrict__ idxs,      // [N_TRUNK, 2]
    const int* __restrict__ types,     // [N_TRUNK]
    const float* __restrict__ b1,      // [4, 256]
    const float* __restrict__ b2,      // [4, 128]
    const float* __restrict__ pk1,     // packed W1
    const float* __restrict__ pk2,     // packed W2
    float* __restrict__ buf,           // [NROOT+N_TRUNK, 128]
    unsigned int* __restrict__ flags)  // [N_TRUNK]
{
    __shared__ float xsh[TWOD];        // gathered parent embeddings
    __shared__ float hsh[TWOD];        // hidden activations
    __shared__ float bsh1[4 * TWOD];   // all b1 biases (4 KB)
    __shared__ float bsh2[4 * DIM];    // all b2 biases (2 KB)

    const int tid  = threadIdx.x;
    const int w    = tid >> 5;        // wave id 0..7
    const int lane = tid & 31;
    const int hi   = lane >> 4;       // selects K pair {2hi, 2hi+1}
    const int loff = lane * 2;        // packed-B per-lane dword offset

    // preload all biases once; reused by every node this workgroup processes
    for (int e = tid; e < 4 * TWOD; e += NTHREADS) bsh1[e] = b1[e];
    for (int e = tid; e < 4 * DIM;  e += NTHREADS) bsh2[e] = b2[e];
    __syncthreads();

    for (int i = blockIdx.x; i < N_TRUNK; i += NWG) {
        const int p0 = idxs[2 * i];
        const int p1 = idxs[2 * i + 1];
        const int t  = types[i];

        // ---- wait until both parents are written; poll them in parallel ----
        if (tid == 0 && p0 >= NROOT)
            while (__hip_atomic_load(&flags[p0 - NROOT], __ATOMIC_RELAXED,
                                     __HIP_MEMORY_SCOPE_AGENT) == 0u)
                __builtin_amdgcn_s_sleep(1);
        if (tid == 32 && p1 >= NROOT)
            while (__hip_atomic_load(&flags[p1 - NROOT], __ATOMIC_RELAXED,
                                     __HIP_MEMORY_SCOPE_AGENT) == 0u)
                __builtin_amdgcn_s_sleep(1);
        __syncthreads();
        __builtin_amdgcn_fence(__ATOMIC_ACQUIRE, "agent");

        // ---- gather parent rows into LDS: x = concat(buf[p0], buf[p1]) ----
        if (tid < DIM) xsh[tid] = buf[(size_t)p0 * DIM + tid];
        else           xsh[tid] = buf[(size_t)p1 * DIM + (tid - DIM)];
        __syncthreads();

        // ---- layer 1: h = gelu(x @ W1[t] + b1[t]); wave w owns cols [32w,32w+32)
        // All lanes load the same x-pair -> every A row is a copy of x; we only
        // read D row M=0, so no masking is needed (values stay finite).
        // 2 accumulators per tile break the C-accumulation RAW chain.
        const float* __restrict__ pA = pk1 + ((size_t)(t * 16 + 2 * w)    ) * (64 * 64);
        const float* __restrict__ pB = pk1 + ((size_t)(t * 16 + 2 * w + 1)) * (64 * 64);
        v8f c0a = {}, c0b = {}, c1a = {}, c1b = {};
        #pragma unroll 4
        for (int ks = 0; ks < 64; ks += 2) {
            v2f a0  = *(const v2f*)(xsh + 4 * ks + 2 * hi);
            v2f a1  = *(const v2f*)(xsh + 4 * ks + 4 + 2 * hi);
            v2f bA0 = *(const v2f*)(pA + ks * 64 + loff);
            v2f bB0 = *(const v2f*)(pB + ks * 64 + loff);
            v2f bA1 = *(const v2f*)(pA + ks * 64 + 64 + loff);
            v2f bB1 = *(const v2f*)(pB + ks * 64 + 64 + loff);
            c0a = __builtin_amdgcn_wmma_f32_16x16x4_f32(false, a0, false, bA0,
                                                        (short)0, c0a, false, false);
            c1a = __builtin_amdgcn_wmma_f32_16x16x4_f32(false, a0, false, bB0,
                                                        (short)0, c1a, false, false);
            c0b = __builtin_amdgcn_wmma_f32_16x16x4_f32(false, a1, false, bA1,
                                                        (short)0, c0b, false, false);
            c1b = __builtin_amdgcn_wmma_f32_16x16x4_f32(false, a1, false, bB1,
                                                        (short)0, c1b, false, false);
        }
        if (lane < 16) {  // row M=0 lives in C-VGPR0 of lanes 0..15
            const float* b1t = bsh1 + t * TWOD + 32 * w + lane;
            hsh[32 * w + lane]      = gelu_exact(c0a[0] + c0b[0] + b1t[0]);
            hsh[32 * w + 16 + lane] = gelu_exact(c1a[0] + c1b[0] + b1t[16]);
        }
        __syncthreads();

        // ---- layer 2: y = h @ W2[t] + b2[t]; wave w owns cols [16w,16w+16)
        // 4 accumulators -> dependent WMMAs spaced 4 apart.
        const float* __restrict__ pC = pk2 + ((size_t)(t * 8 + w)) * (64 * 64);
        v8f cy0 = {}, cy1 = {}, cy2 = {}, cy3 = {};
        #pragma unroll 2
        for (int ks = 0; ks < 64; ks += 4) {
            v2f a0  = *(const v2f*)(hsh + 4 * ks + 2 * hi);
            v2f a1  = *(const v2f*)(hsh + 4 * ks + 4 + 2 * hi);
            v2f a2  = *(const v2f*)(hsh + 4 * ks + 8 + 2 * hi);
            v2f a3  = *(const v2f*)(hsh + 4 * ks + 12 + 2 * hi);
            v2f b0  = *(const v2f*)(pC + ks * 64 + loff);
            v2f b1v = *(const v2f*)(pC + ks * 64 + 64 + loff);
            v2f b2v = *(const v2f*)(pC + ks * 64 + 128 + loff);
            v2f b3v = *(const v2f*)(pC + ks * 64 + 192 + loff);
            cy0 = __builtin_amdgcn_wmma_f32_16x16x4_f32(false, a0, false, b0,
                                                        (short)0, cy0, false, false);
            cy1 = __builtin_amdgcn_wmma_f32_16x16x4_f32(false, a1, false, b1v,
                                                        (short)0, cy1, false, false);
            cy2 = __builtin_amdgcn_wmma_f32_16x16x4_f32(false, a2, false, b2v,
                                                        (short)0, cy2, false, false);
            cy3 = __builtin_amdgcn_wmma_f32_16x16x4_f32(false, a3, false, b3v,
                                                        (short)0, cy3, false, false);
        }
        if (lane < 16) {
            buf[(size_t)(NROOT + i) * DIM + 16 * w + lane] =
                (cy0[0] + cy1[0]) + (cy2[0] + cy3[0]) + bsh2[t * DIM + 16 * w + lane];
        }

        // ---- publish: drain stores + writeback, then set done flag ----
        __builtin_amdgcn_fence(__ATOMIC_RELEASE, "agent");
        __syncthreads();
        if (tid == 0)
            __hip_atomic_store(&flags[i], 1u, __ATOMIC_RELAXED,
                               __HIP_MEMORY_SCOPE_AGENT);
    }
}

extern "C" void kernel_launch(void* const* d_in, const int* in_sizes, int n_in,
                              void* d_out, int out_size, void* d_ws, size_t ws_size,
                              hipStream_t stream) {
    const int*   idxs  = (const int*)d_in[0];   // trunk_node_inputs_indices
    const int*   types = (const int*)d_in[1];   // trunk_node_types
    const float* roots = (const float*)d_in[2]; // root_node_embeddings
    const float* W1    = (const float*)d_in[3];
    const float* b1    = (const float*)d_in[4];
    const float* W2    = (const float*)d_in[5];
    const float* b2    = (const float*)d_in[6];
    float* out = (float*)d_out;                 // [65600, 128] f32

    unsigned int* flags = (unsigned int*)d_ws;
    float* pk1 = (float*)((char*)d_ws + FLAGS_BYTES);
    float* pk2 = pk1 + PK1_ELEMS;

    // init covers: 65536 flag clears, 8192 root copies, 262144 + 131072 packs
    dag_init<<<(PK1_ELEMS + 255) / 256, 256, 0, stream>>>(roots, W1, W2, out,
                                                          flags, pk1, pk2);
    dag_mlp<<<NWG, NTHREADS, 0, stream>>>(idxs, types, b1, b2, pk1, pk2, out, flags);
}